// Net_MDA_18975165514166
// MI455X (gfx1250) — compile-verified
//
#include <hip/hip_runtime.h>

#define Bn 16
#define Np 1024
#define Kk 20
#define MAXO 1024

typedef __attribute__((ext_vector_type(2))) float v2f;
typedef __attribute__((ext_vector_type(8))) float v8f;

__device__ __forceinline__ float actf(float x, int act) {
  if (act == 1) return x > 0.f ? x : 0.2f * x;   // leaky relu 0.2
  if (act == 2) return fmaxf(x, 0.f);            // relu
  return x;
}

__device__ __forceinline__ void atomicMaxF(float* addr, float val) {
  unsigned int* ua = (unsigned int*)addr;
  unsigned int cur = *ua;
  while (__uint_as_float(cur) < val) {
    unsigned int prev = atomicCAS(ua, cur, __float_as_uint(val));
    if (prev == cur) break;
    cur = prev;
  }
}

__global__ void fill_kernel(float* p, float v, long n) {
  long t = (long)blockIdx.x * blockDim.x + threadIdx.x;
  if (t < n) p[t] = v;
}

__global__ void finalize_stats(const float* __restrict__ stats, float* __restrict__ mnr,
                               int O, float invc) {
  int o = blockIdx.x * blockDim.x + threadIdx.x;
  if (o >= O) return;
  float mean = stats[o] * invc;
  float var  = stats[MAXO + o] * invc - mean * mean;
  mnr[o] = mean;
  mnr[MAXO + o] = rsqrtf(var + 1e-5f);
}

// ---------------- KNN top-k (matches jax.lax.top_k tie-break: lowest index) --------------
__global__ void knn_kernel(const float* __restrict__ pts, long pbs, long pcs, int Npts,
                           const float* __restrict__ qry, long qbs, long qcs, int Mq,
                           int C, int k, int* __restrict__ idx_out, float* __restrict__ d_out) {
  int b = blockIdx.x / Mq;
  int m = blockIdx.x % Mq;
  __shared__ float qv[128];
  __shared__ float negd[1024];
  __shared__ float rv[256];
  __shared__ int   ri[256];
  int t = threadIdx.x;
  for (int c = t; c < C; c += 256) qv[c] = qry[(long)b * qbs + (long)c * qcs + m];
  __syncthreads();
  float qsq = 0.f;
  for (int c = 0; c < C; ++c) qsq += qv[c] * qv[c];
  for (int j = t; j < Npts; j += 256) {
    float dot = 0.f, psq = 0.f;
    for (int c = 0; c < C; ++c) {
      float pv = pts[(long)b * pbs + (long)c * pcs + j];
      dot += qv[c] * pv; psq += pv * pv;
    }
    negd[j] = 2.f * dot - qsq - psq;
  }
  __syncthreads();
  for (int s = 0; s < k; ++s) {
    float bv = -3.4e38f; int bi = 0x7fffffff;
    for (int j = t; j < Npts; j += 256) {
      float v = negd[j];
      if (v > bv || (v == bv && j < bi)) { bv = v; bi = j; }
    }
    rv[t] = bv; ri[t] = bi;
    __syncthreads();
    for (int off = 128; off > 0; off >>= 1) {
      if (t < off) {
        if (rv[t + off] > rv[t] || (rv[t + off] == rv[t] && ri[t + off] < ri[t])) {
          rv[t] = rv[t + off]; ri[t] = ri[t + off];
        }
      }
      __syncthreads();
    }
    if (t == 0) {
      int bidx = ri[0];
      idx_out[((long)b * Mq + m) * k + s] = bidx;
      if (d_out) d_out[((long)b * Mq + m) * k + s] = -rv[0];  // positive squared distance
      negd[bidx] = -3.4e38f;
    }
    __syncthreads();
  }
}

// ---------------- Farthest point sampling (NUM_NODE = 64) ----------------
__global__ void fps_kernel(const float* __restrict__ loc, int* __restrict__ fidx) {
  int b = blockIdx.x;
  __shared__ float dmin[1024];
  __shared__ float rv[256];
  __shared__ int   ri[256];
  __shared__ int lastS;
  int t = threadIdx.x;
  for (int j = t; j < Np; j += 256) dmin[j] = 1e10f;
  if (t == 0) { fidx[b * 64] = 0; lastS = 0; }
  __syncthreads();
  for (int s = 1; s < 64; ++s) {
    int last = lastS;
    float lx = loc[((long)b * 3 + 0) * Np + last];
    float ly = loc[((long)b * 3 + 1) * Np + last];
    float lz = loc[((long)b * 3 + 2) * Np + last];
    float bv = -3.4e38f; int bi = 0x7fffffff;
    for (int j = t; j < Np; j += 256) {
      float dx = loc[((long)b * 3 + 0) * Np + j] - lx;
      float dy = loc[((long)b * 3 + 1) * Np + j] - ly;
      float dz = loc[((long)b * 3 + 2) * Np + j] - lz;
      float nd = dx * dx + dy * dy + dz * dz;
      float d = fminf(dmin[j], nd);
      dmin[j] = d;
      if (d > bv || (d == bv && j < bi)) { bv = d; bi = j; }
    }
    rv[t] = bv; ri[t] = bi;
    __syncthreads();
    for (int off = 128; off > 0; off >>= 1) {
      if (t < off) {
        if (rv[t + off] > rv[t] || (rv[t + off] == rv[t] && ri[t + off] < ri[t])) {
          rv[t] = rv[t + off]; ri[t] = ri[t + off];
        }
      }
      __syncthreads();
    }
    if (t == 0) { fidx[b * 64 + s] = ri[0]; lastS = ri[0]; }
    __syncthreads();
  }
}

// =====================================================================================
// WMMA GEMM on graph features built on the fly.
// y[o, j] = sum_c W[o,c] * F[c,j];  j=(b,n,kk);  F = [nbr - ctr ; ctr], C2 = 2*Cin
// PRECONDITION: (Bn*N*kK) % 16 == 0  (true for all launches here)
// MODE 0: accumulate per-channel sum / sumsq into stats.  MODE 1: BN+act, write/atomic-max.
// =====================================================================================
template <int MODE>
__global__ void gff_gemm(const float* __restrict__ W, int O, int Cin,
                         const float* __restrict__ X, long xbs, long xcs,
                         const int* __restrict__ idx, int N, int kK,
                         float* __restrict__ stats, const float* __restrict__ mnr,
                         int act, float* __restrict__ out, long obs, long ocs, int ooff,
                         int maxdiv, int use_amax) {
  const int C2 = 2 * Cin;
  const int NK = N * kK;
  const int J = Bn * NK;
  const int Jt = J >> 4;
  const int lane = threadIdx.x;
  const int mrow = lane & 15;
  const int g = lane >> 4;
  const int o0 = blockIdx.x << 4;
  const int cfull = C2 >> 2;            // guard-free K steps
  const int has_tail = (C2 & 3) != 0;
  const float* __restrict__ Wrow = W + (long)(o0 + mrow) * C2;

  float sA[8] = {0.f, 0.f, 0.f, 0.f, 0.f, 0.f, 0.f, 0.f};
  float sQ[8] = {0.f, 0.f, 0.f, 0.f, 0.f, 0.f, 0.f, 0.f};
  float mean[8], rstd[8];
  if (MODE == 1) {
#pragma unroll
    for (int r = 0; r < 8; ++r) {
      int o = o0 + r + (g << 3);
      mean[r] = mnr[o];
      rstd[r] = mnr[MAXO + o];
    }
  }

  for (int jt = blockIdx.y; jt < Jt; jt += gridDim.y) {
    int j = (jt << 4) + mrow;           // this lane's column (always < J)
    int bb = j / NK;
    int rem = j - bb * NK;
    int nn = rem / kK;
    int kkidx = rem - nn * kK;
    int nbr = idx[(bb * N + nn) * kK + kkidx];
    const float* __restrict__ Pc = X + (long)bb * xbs + nn;   // center column
    const float* __restrict__ Pn = X + (long)bb * xbs + nbr;  // neighbor column
    v8f acc = {};
    for (int cs = 0; cs < cfull; ++cs) {
      int ca = (cs << 2) + (g << 1);
      int cb = ca + 1;
      int ca2 = ca < Cin ? ca : ca - Cin;   // always in [0, Cin)
      int cb2 = cb < Cin ? cb : cb - Cin;
      v2f av, bv;
      av[0] = Wrow[ca];
      av[1] = Wrow[cb];
      float c0 = Pc[(long)ca2 * xcs];
      float c1 = Pc[(long)cb2 * xcs];
      float n0 = Pn[(long)ca2 * xcs];
      float n1 = Pn[(long)cb2 * xcs];
      bv[0] = ca < Cin ? n0 - c0 : c0;
      bv[1] = cb < Cin ? n1 - c1 : c1;
      acc = __builtin_amdgcn_wmma_f32_16x16x4_f32(false, av, false, bv, (short)0, acc,
                                                  false, false);
    }
    if (has_tail) {                      // only C2 % 4 != 0 (the 6-channel convs)
      int ca = (cfull << 2) + (g << 1);
      int cb = ca + 1;
      bool va = ca < C2, vb = cb < C2;
      int ca2 = ca < Cin ? ca : (va ? ca - Cin : 0);
      int cb2 = cb < Cin ? cb : (vb ? cb - Cin : 0);
      v2f av, bv;
      av[0] = va ? Wrow[ca] : 0.f;
      av[1] = vb ? Wrow[cb] : 0.f;
      float c0 = Pc[(long)ca2 * xcs];
      float c1 = Pc[(long)cb2 * xcs];
      float n0 = Pn[(long)ca2 * xcs];
      float n1 = Pn[(long)cb2 * xcs];
      bv[0] = !va ? 0.f : (ca < Cin ? n0 - c0 : c0);
      bv[1] = !vb ? 0.f : (cb < Cin ? n1 - c1 : c1);
      acc = __builtin_amdgcn_wmma_f32_16x16x4_f32(false, av, false, bv, (short)0, acc,
                                                  false, false);
    }
    if (MODE == 0) {
#pragma unroll
      for (int r = 0; r < 8; ++r) { float y = acc[r]; sA[r] += y; sQ[r] += y * y; }
    } else {
      int outn = rem / maxdiv;
#pragma unroll
      for (int r = 0; r < 8; ++r) {
        int o = o0 + r + (g << 3);
        float yn = actf((acc[r] - mean[r]) * rstd[r], act);
        long dst = (long)bb * obs + (long)(ooff + o) * ocs + outn;
        if (use_amax) atomicMaxF(out + dst, yn);
        else out[dst] = yn;
      }
    }
  }
  if (MODE == 0) {
#pragma unroll
    for (int r = 0; r < 8; ++r) {
      float s = sA[r], q = sQ[r];
      for (int msk = 1; msk < 16; msk <<= 1) {
        s += __shfl_xor(s, msk, 32);
        q += __shfl_xor(q, msk, 32);
      }
      if (mrow == 0) {
        int o = o0 + r + (g << 3);
        atomicAdd(&stats[o], s);
        atomicAdd(&stats[MAXO + o], q);
      }
    }
  }
}

// =====================================================================================
// Dense WMMA GEMM with strided input.  F[c, j] = X[(j/N)*xbs + c*xcs + (j%N)]
// PRECONDITION: (Bn*N) % 16 == 0 and C % 4 == 0 (true for all launches here)
// =====================================================================================
template <int MODE>
__global__ void gemm_bn(const float* __restrict__ W, int O, int C,
                        const float* __restrict__ X, long xbs, long xcs, int N,
                        const float* __restrict__ bias,
                        float* __restrict__ stats, const float* __restrict__ mnr,
                        int act, float* __restrict__ out, long obs, long ocs,
                        int maxdiv, int use_amax) {
  const int J = Bn * N;
  const int Jt = J >> 4;
  const int lane = threadIdx.x;
  const int mrow = lane & 15;
  const int g = lane >> 4;
  const int o0 = blockIdx.x << 4;
  const int csteps = C >> 2;
  const float* __restrict__ Wrow = W + (long)(o0 + mrow) * C;

  float sA[8] = {0.f, 0.f, 0.f, 0.f, 0.f, 0.f, 0.f, 0.f};
  float sQ[8] = {0.f, 0.f, 0.f, 0.f, 0.f, 0.f, 0.f, 0.f};
  float bval[8], mean[8], rstd[8];
#pragma unroll
  for (int r = 0; r < 8; ++r) {
    int o = o0 + r + (g << 3);
    bval[r] = bias ? bias[o] : 0.f;
    if (MODE == 1) { mean[r] = mnr[o]; rstd[r] = mnr[MAXO + o]; }
  }

  for (int jt = blockIdx.y; jt < Jt; jt += gridDim.y) {
    int j = (jt << 4) + mrow;
    int bb = j / N;
    int nn = j - bb * N;
    const float* __restrict__ Pc = X + (long)bb * xbs + nn;
    v8f acc = {};
    for (int cs = 0; cs < csteps; ++cs) {
      int ca = (cs << 2) + (g << 1);
      v2f av, bv;
      av[0] = Wrow[ca];
      av[1] = Wrow[ca + 1];
      bv[0] = Pc[(long)ca * xcs];
      bv[1] = Pc[(long)(ca + 1) * xcs];
      acc = __builtin_amdgcn_wmma_f32_16x16x4_f32(false, av, false, bv, (short)0, acc,
                                                  false, false);
    }
    if (MODE == 0) {
#pragma unroll
      for (int r = 0; r < 8; ++r) {
        float y = acc[r] + bval[r];
        sA[r] += y; sQ[r] += y * y;
      }
    } else {
      int outn = nn / maxdiv;
#pragma unroll
      for (int r = 0; r < 8; ++r) {
        int o = o0 + r + (g << 3);
        float yn = actf((acc[r] + bval[r] - mean[r]) * rstd[r], act);
        long dst = (long)bb * obs + (long)o * ocs + outn;
        if (use_amax) atomicMaxF(out + dst, yn);
        else out[dst] = yn;
      }
    }
  }
  if (MODE == 0) {
#pragma unroll
    for (int r = 0; r < 8; ++r) {
      float s = sA[r], q = sQ[r];
      for (int msk = 1; msk < 16; msk <<= 1) {
        s += __shfl_xor(s, msk, 32);
        q += __shfl_xor(q, msk, 32);
      }
      if (mrow == 0) {
        int o = o0 + r + (g << 3);
        atomicAdd(&stats[o], s);
        atomicAdd(&stats[MAXO + o], q);
      }
    }
  }
}

// ---------------- misc small kernels ----------------
__global__ void gather_kernel(const float* __restrict__ in, long ibs, long ics, int C,
                              const int* __restrict__ gi, int M, float* __restrict__ out) {
  long tot = (long)Bn * C * M;
  for (long t = (long)blockIdx.x * blockDim.x + threadIdx.x; t < tot;
       t += (long)gridDim.x * blockDim.x) {
    int m = (int)(t % M);
    long r = t / M;
    int c = (int)(r % C);
    int b = (int)(r / C);
    out[((long)b * C + c) * M + m] = in[(long)b * ibs + (long)c * ics + gi[b * M + m]];
  }
}

__global__ void small_fc(const float* __restrict__ X, const float* __restrict__ W,
                         const float* __restrict__ bias, int C, int O,
                         float* __restrict__ out, int add_eye) {
  int t = blockIdx.x * blockDim.x + threadIdx.x;
  if (t >= Bn * O) return;
  int b = t / O, o = t % O;
  float acc = bias ? bias[o] : 0.f;
  const float* xr = X + (long)b * C;
  const float* wr = W + (long)o * C;
  for (int c = 0; c < C; ++c) acc += xr[c] * wr[c];
  if (add_eye) { int i = o / 3, jj = o % 3; if (i == jj) acc += 1.f; }
  out[t] = acc;
}

__global__ void tapply_kernel(const float* __restrict__ T, const float* __restrict__ x,
                              float* __restrict__ xt) {
  long t = (long)blockIdx.x * blockDim.x + threadIdx.x;
  if (t >= (long)Bn * 3 * Np) return;
  int n = (int)(t % Np);
  long r = t / Np;
  int i = (int)(r % 3);
  int b = (int)(r / 3);
  const float* Tb = T + b * 9 + i * 3;
  float v = Tb[0] * x[((long)b * 3 + 0) * Np + n] +
            Tb[1] * x[((long)b * 3 + 1) * Np + n] +
            Tb[2] * x[((long)b * 3 + 2) * Np + n];
  xt[t] = v;
}

// node offsets: tanh(off_w @ (fea[gidx]-ffea)) * (loc[gidx]-floc), mean over 64 -> node_loc
__global__ void node_off_kernel(const float* __restrict__ offW, const float* __restrict__ fea,
                                const float* __restrict__ loc, const float* __restrict__ floc,
                                const float* __restrict__ ffea, const int* __restrict__ gidx,
                                float* __restrict__ nloc) {
  int b = blockIdx.x / 64, m = blockIdx.x % 64;
  int kk = threadIdx.x;  // 64 threads
  int i = gidx[((long)b * 64 + m) * 64 + kk];
  float d0 = 0.f, d1 = 0.f, d2 = 0.f;
  for (int c = 0; c < 512; ++c) {
    float f = fea[((long)b * 512 + c) * Np + i] - ffea[((long)b * 512 + c) * 64 + m];
    d0 += offW[c] * f;
    d1 += offW[512 + c] * f;
    d2 += offW[1024 + c] * f;
  }
  __shared__ float red0[64], red1[64], red2[64];
  float g0 = loc[((long)b * 3 + 0) * Np + i] - floc[((long)b * 3 + 0) * 64 + m];
  float g1 = loc[((long)b * 3 + 1) * Np + i] - floc[((long)b * 3 + 1) * 64 + m];
  float g2 = loc[((long)b * 3 + 2) * Np + i] - floc[((long)b * 3 + 2) * 64 + m];
  red0[kk] = tanhf(d0) * g0;
  red1[kk] = tanhf(d1) * g1;
  red2[kk] = tanhf(d2) * g2;
  __syncthreads();
  for (int off = 32; off > 0; off >>= 1) {
    if (kk < off) { red0[kk] += red0[kk + off]; red1[kk] += red1[kk + off]; red2[kk] += red2[kk + off]; }
    __syncthreads();
  }
  if (kk == 0) {
    nloc[((long)b * 3 + 0) * 64 + m] = floc[((long)b * 3 + 0) * 64 + m] + red0[0] * (1.f / 64.f);
    nloc[((long)b * 3 + 1) * 64 + m] = floc[((long)b * 3 + 1) * 64 + m] + red1[0] * (1.f / 64.f);
    nloc[((long)b * 3 + 2) * 64 + m] = floc[((long)b * 3 + 2) * 64 + m] + red2[0] * (1.f / 64.f);
  }
}

__global__ void node_pool_kernel(const float* __restrict__ fea, const int* __restrict__ gidx2,
                                 float* __restrict__ npool) {
  int b = blockIdx.x / 64, m = blockIdx.x % 64;
  __shared__ int ids[64];
  if (threadIdx.x < 64) ids[threadIdx.x] = gidx2[((long)b * 64 + m) * 64 + threadIdx.x];
  __syncthreads();
  for (int c = threadIdx.x; c < 512; c += blockDim.x) {
    const float* fr = fea + ((long)b * 512 + c) * Np;
    float mx = -3.4e38f;
    for (int kk = 0; kk < 64; ++kk) mx = fmaxf(mx, fr[ids[kk]]);
    npool[((long)b * 512 + c) * 64 + m] = mx;
  }
}

__global__ void interp_kernel(const float* __restrict__ fea, const float* __restrict__ npool,
                              const int* __restrict__ ni, const float* __restrict__ nd,
                              float* __restrict__ xfea) {
  int b = blockIdx.x / Np, n = blockIdx.x % Np;
  long base = ((long)b * Np + n) * 3;
  int i0 = ni[base + 0], i1 = ni[base + 1], i2 = ni[base + 2];
  float w0 = 1.f / (fmaxf(nd[base + 0], 0.f) + 1e-8f);
  float w1 = 1.f / (fmaxf(nd[base + 1], 0.f) + 1e-8f);
  float w2 = 1.f / (fmaxf(nd[base + 2], 0.f) + 1e-8f);
  float inv = 1.f / (w0 + w1 + w2);
  w0 *= inv; w1 *= inv; w2 *= inv;
  for (int c = threadIdx.x; c < 512; c += blockDim.x) {
    long fc = (long)b * 512 + c;
    float v = fea[fc * Np + n];
    v += w0 * npool[fc * 64 + i0] + w1 * npool[fc * 64 + i1] + w2 * npool[fc * 64 + i2];
    xfea[fc * Np + n] = v;
  }
}

// ---------------- host-side helpers ----------------
static void run_gff(hipStream_t s, const float* Wm, int O, int Cin,
                    const float* X, long xbs, long xcs, const int* idx, int N, int kK,
                    float* STATS, float* MNR, int act,
                    float* out, long obs, long ocs, int ooff, int maxdiv, int amax) {
  long J = (long)Bn * N * kK;
  long Jt = J >> 4;
  int gy = (int)(Jt < 1024 ? Jt : 1024);
  fill_kernel<<<8, 256, 0, s>>>(STATS, 0.f, 2048);
  dim3 grid((unsigned)(O / 16), (unsigned)gy);
  gff_gemm<0><<<grid, 32, 0, s>>>(Wm, O, Cin, X, xbs, xcs, idx, N, kK, STATS, MNR, 0,
                                  nullptr, 0, 0, 0, 1, 0);
  finalize_stats<<<(O + 63) / 64, 64, 0, s>>>(STATS, MNR, O, 1.f / (float)J);
  gff_gemm<1><<<grid, 32, 0, s>>>(Wm, O, Cin, X, xbs, xcs, idx, N, kK, STATS, MNR, act,
                                  out, obs, ocs, ooff, maxdiv, amax);
}

static void run_gemm(hipStream_t s, const float* Wm, int O, int C,
                     const float* X, long xbs, long xcs, int N, const float* bias,
                     float* STATS, float* MNR, int act,
                     float* out, long obs, long ocs, int maxdiv, int amax) {
  long J = (long)Bn * N;
  long Jt = J >> 4;
  int gy = (int)(Jt < 1024 ? Jt : 1024);
  if (gy < 1) gy = 1;
  fill_kernel<<<8, 256, 0, s>>>(STATS, 0.f, 2048);
  dim3 grid((unsigned)(O / 16), (unsigned)gy);
  gemm_bn<0><<<grid, 32, 0, s>>>(Wm, O, C, X, xbs, xcs, N, bias, STATS, MNR, 0,
                                 nullptr, 0, 0, 1, 0);
  finalize_stats<<<(O + 63) / 64, 64, 0, s>>>(STATS, MNR, O, 1.f / (float)J);
  gemm_bn<1><<<grid, 32, 0, s>>>(Wm, O, C, X, xbs, xcs, N, bias, STATS, MNR, act,
                                 out, obs, ocs, maxdiv, amax);
}

extern "C" void kernel_launch(void* const* d_in, const int* in_sizes, int n_in,
                              void* d_out, int out_size, void* d_ws, size_t ws_size,
                              hipStream_t stream) {
  (void)in_sizes; (void)n_in; (void)out_size; (void)ws_size;
  hipStream_t s = stream;
  const float* x      = (const float*)d_in[0];   // (16,3,1024,1) == (16,3,1024)
  const float* P_tc1  = (const float*)d_in[1];   // (64,6)
  const float* P_tc2  = (const float*)d_in[2];   // (128,64)
  const float* P_tc3  = (const float*)d_in[3];   // (1024,128)
  const float* P_tfc1 = (const float*)d_in[4];   // (512,1024)
  const float* P_tfc2 = (const float*)d_in[5];   // (256,512)
  const float* P_tfc2b= (const float*)d_in[6];
  const float* P_tfc3 = (const float*)d_in[7];   // (9,256)
  const float* P_tfc3b= (const float*)d_in[8];
  const float* P_c1   = (const float*)d_in[9];   // (64,6)
  const float* P_c2   = (const float*)d_in[10];  // (64,128)
  const float* P_c3   = (const float*)d_in[11];  // (128,128)
  const float* P_c4   = (const float*)d_in[12];  // (256,256)
  const float* P_c6   = (const float*)d_in[13];  // (1024,512)
  const float* P_off  = (const float*)d_in[14];  // (3,512)
  // d_in[15..18] trans_w/trans_b/res_w/res_b: dead for y1/y2
  const float* C1w1 = (const float*)d_in[19]; const float* C1b1 = (const float*)d_in[20];
  const float* C1w2 = (const float*)d_in[21]; const float* C1b2 = (const float*)d_in[22];
  const float* C1w3 = (const float*)d_in[23]; const float* C1b3 = (const float*)d_in[24];
  const float* C2w1 = (const float*)d_in[25]; const float* C2b1 = (const float*)d_in[26];
  const float* C2w2 = (const float*)d_in[27]; const float* C2b2 = (const float*)d_in[28];
  const float* C2w3 = (const float*)d_in[29]; const float* C2b3 = (const float*)d_in[30];

  // ---- workspace layout (floats) ----
  float* wsf  = (float*)d_ws;
  float* STATS = wsf;                       // 2048
  float* MNR   = wsf + 2048;                // 2048
  int*   IDX20 = (int*)(wsf + 4096);        // 16*1024*20 = 327680
  int*   FIDX  = IDX20 + 327680;            // 1024
  int*   GIDX  = FIDX + 1024;               // 65536
  int*   GIDX2 = GIDX + 65536;              // 65536
  int*   NI    = GIDX2 + 65536;             // 49152
  float* ND    = (float*)(NI + 49152);      // 49152
  float* T1    = ND + 49152;                // 16*64*20480 = 20971520
  float* X128  = T1 + 20971520;             // 16*128*1024 = 2097152
  float* TG    = X128 + 2097152;            // 16384
  float* FC1   = TG + 16384;                // 8192
  float* FC2   = FC1 + 8192;                // 4096
  float* TMAT  = FC2 + 4096;                // 144
  float* XT    = TMAT + 144;                // 49152
  float* XCAT  = XT + 49152;                // 16*512*1024 = 8388608
  float* FLOC  = XCAT + 8388608;            // 3072
  float* FFEA  = FLOC + 3072;               // 524288
  float* NLOC  = FFEA + 524288;             // 3072
  float* NPOOL = NLOC + 3072;               // 524288
  float* XFEA  = NPOOL + 524288;            // 8388608
  float* X6    = XFEA + 8388608;            // 16384
  float* H1    = X6 + 16384;                // 8192
  float* H2    = H1 + 8192;                 // 4096

  // ================= transform net =================
  knn_kernel<<<Bn * Np, 256, 0, s>>>(x, 3L * Np, Np, Np, x, 3L * Np, Np, Np, 3, Kk, IDX20, nullptr);
  // t_conv1: graph feature (6ch) -> 64ch, full write to T1 (B,64,N*K)
  run_gff(s, P_tc1, 64, 3, x, 3L * Np, Np, IDX20, Np, Kk, STATS, MNR, 1,
          T1, 64L * 20480, 20480, 0, 1, 0);
  // t_conv2: 64 -> 128, bn+lrelu, max over k -> X128 (B,128,N)
  fill_kernel<<<(2097152 + 255) / 256, 256, 0, s>>>(X128, -3.0e38f, 2097152);
  run_gemm(s, P_tc2, 128, 64, T1, 64L * 20480, 20480, 20480, nullptr, STATS, MNR, 1,
           X128, 128L * Np, Np, Kk, 1);
  // t_conv3: 128 -> 1024, bn+lrelu, max over n -> TG (B,1024)
  fill_kernel<<<64, 256, 0, s>>>(TG, -3.0e38f, 16384);
  run_gemm(s, P_tc3, 1024, 128, X128, 128L * Np, Np, Np, nullptr, STATS, MNR, 1,
           TG, 1024, 1, Np, 1);
  // fc1 / fc2 (row-major (16,C) inputs via xbs=C, xcs=1, N=1)
  run_gemm(s, P_tfc1, 512, 1024, TG, 1024, 1, 1, nullptr, STATS, MNR, 1, FC1, 512, 1, 1, 0);
  run_gemm(s, P_tfc2, 256, 512, FC1, 512, 1, 1, P_tfc2b, STATS, MNR, 1, FC2, 256, 1, 1, 0);
  small_fc<<<1, 256, 0, s>>>(FC2, P_tfc3, P_tfc3b, 256, 9, TMAT, 1);   // T = fc3 + I
  tapply_kernel<<<(Bn * 3 * Np + 255) / 256, 256, 0, s>>>(TMAT, x, XT);

  // ================= edge conv stack (outputs concatenated into XCAT) =================
  fill_kernel<<<(8388608 + 255) / 256, 256, 0, s>>>(XCAT, -3.0e38f, 8388608);
  knn_kernel<<<Bn * Np, 256, 0, s>>>(XT, 3L * Np, Np, Np, XT, 3L * Np, Np, Np, 3, Kk, IDX20, nullptr);
  run_gff(s, P_c1, 64, 3, XT, 3L * Np, Np, IDX20, Np, Kk, STATS, MNR, 1,
          XCAT, 512L * Np, Np, 0, Kk, 1);
  knn_kernel<<<Bn * Np, 256, 0, s>>>(XCAT, 512L * Np, Np, Np, XCAT, 512L * Np, Np, Np, 64, Kk, IDX20, nullptr);
  run_gff(s, P_c2, 64, 64, XCAT, 512L * Np, Np, IDX20, Np, Kk, STATS, MNR, 1,
          XCAT, 512L * Np, Np, 64, Kk, 1);
  knn_kernel<<<Bn * Np, 256, 0, s>>>(XCAT + 64L * Np, 512L * Np, Np, Np,
                                     XCAT + 64L * Np, 512L * Np, Np, Np, 64, Kk, IDX20, nullptr);
  run_gff(s, P_c3, 128, 64, XCAT + 64L * Np, 512L * Np, Np, IDX20, Np, Kk, STATS, MNR, 1,
          XCAT, 512L * Np, Np, 128, Kk, 1);
  knn_kernel<<<Bn * Np, 256, 0, s>>>(XCAT + 128L * Np, 512L * Np, Np, Np,
                                     XCAT + 128L * Np, 512L * Np, Np, Np, 128, Kk, IDX20, nullptr);
  run_gff(s, P_c4, 256, 128, XCAT + 128L * Np, 512L * Np, Np, IDX20, Np, Kk, STATS, MNR, 1,
          XCAT, 512L * Np, Np, 256, Kk, 1);

  // ================= adapt layer (only x_fea path matters) =================
  fps_kernel<<<Bn, 256, 0, s>>>(x, FIDX);
  gather_kernel<<<(Bn * 3 * 64 + 255) / 256, 256, 0, s>>>(x, 3L * Np, Np, 3, FIDX, 64, FLOC);
  gather_kernel<<<(Bn * 512 * 64 + 255) / 256, 256, 0, s>>>(XCAT, 512L * Np, Np, 512, FIDX, 64, FFEA);
  knn_kernel<<<Bn * 64, 256, 0, s>>>(x, 3L * Np, Np, Np, FLOC, 3L * 64, 64, 64, 3, 64, GIDX, nullptr);
  node_off_kernel<<<Bn * 64, 64, 0, s>>>(P_off, XCAT, x, FLOC, FFEA, GIDX, NLOC);
  knn_kernel<<<Bn * 64, 256, 0, s>>>(x, 3L * Np, Np, Np, NLOC, 3L * 64, 64, 64, 3, 64, GIDX2, nullptr);
  node_pool_kernel<<<Bn * 64, 256, 0, s>>>(XCAT, GIDX2, NPOOL);
  knn_kernel<<<Bn * Np, 256, 0, s>>>(NLOC, 3L * 64, 64, 64, x, 3L * Np, Np, Np, 3, 3, NI, ND);
  interp_kernel<<<Bn * Np, 256, 0, s>>>(XCAT, NPOOL, NI, ND, XFEA);

  // ================= conv6 + global max =================
  fill_kernel<<<64, 256, 0, s>>>(X6, -3.0e38f, 16384);
  run_gemm(s, P_c6, 1024, 512, XFEA, 512L * Np, Np, Np, nullptr, STATS, MNR, 1,
           X6, 1024, 1, Np, 1);

  // ================= classifier heads =================
  run_gemm(s, C1w1, 512, 1024, X6, 1024, 1, 1, C1b1, STATS, MNR, 1, H1, 512, 1, 1, 0);
  run_gemm(s, C1w2, 256, 512, H1, 512, 1, 1, C1b2, STATS, MNR, 1, H2, 256, 1, 1, 0);
  small_fc<<<1, 256, 0, s>>>(H2, C1w3, C1b3, 256, 10, (float*)d_out, 0);
  run_gemm(s, C2w1, 512, 1024, X6, 1024, 1, 1, C2b1, STATS, MNR, 1, H1, 512, 1, 1, 0);
  run_gemm(s, C2w2, 256, 512, H1, 512, 1, 1, C2b2, STATS, MNR, 1, H2, 256, 1, 1, 0);
  small_fc<<<1, 256, 0, s>>>(H2, C2w3, C2b3, 256, 10, (float*)d_out + 160, 0);
}